// HybridQuantumClassifier_2937757630600
// MI455X (gfx1250) — compile-verified
//
#include <hip/hip_runtime.h>
#include <hip/hip_bf16.h>

typedef __attribute__((ext_vector_type(2))) float v2f;
typedef __attribute__((ext_vector_type(4))) float v4f;
typedef __attribute__((ext_vector_type(8))) float v8f;

#define N_QUBITS 12
#define N_LAYERS 6
#define N_CLASSES 10
#define NSTATE 4096   // 2^12
#define NMAT 18       // 6 layers * 3 groups

// ---------------------------------------------------------------------------
// Kernel 1: build the 18 fused 16x16 complex gate matrices (batch independent).
// G[l][g] = U[l][4g] (x) U[l][4g+1] (x) U[l][4g+2] (x) U[l][4g+3]
// Rot(phi,theta,omega) = RZ(omega) RY(theta) RZ(phi); stored row-major [i][k],
// real plane then imag plane.
// ---------------------------------------------------------------------------
__global__ void prep_gates(const float* __restrict__ qp,
                           float* __restrict__ Gr, float* __restrict__ Gi) {
    __shared__ float ur[72][2][2];
    __shared__ float ui[72][2][2];
    int t = threadIdx.x;
    if (t < 72) {
        int l = t / 12, q = t % 12;
        const float* p = qp + (l * 12 + q) * 3;
        float phi = p[0], th = p[1], om = p[2];
        float ct = cosf(0.5f * th), st = sinf(0.5f * th);
        float a = 0.5f * (phi + om), b = 0.5f * (phi - om);
        float ca = cosf(a), sa = sinf(a), cb = cosf(b), sb = sinf(b);
        ur[t][0][0] = ca * ct;  ui[t][0][0] = -sa * ct;   // e^{-ia} c
        ur[t][0][1] = -cb * st; ui[t][0][1] = -sb * st;   // -conj(e^{-ib}) s
        ur[t][1][0] = cb * st;  ui[t][1][0] = -sb * st;   // e^{-ib} s
        ur[t][1][1] = ca * ct;  ui[t][1][1] = sa * ct;    // conj(e^{-ia}) c
    }
    __syncthreads();
    for (int e = t; e < NMAT * 256; e += blockDim.x) {
        int m = e >> 8, pos = e & 255;
        int l = m / 3, g = m % 3;
        int i = pos >> 4, k = pos & 15;
        float pr = 1.f, pi = 0.f;
#pragma unroll
        for (int j = 0; j < 4; ++j) {
            int u = l * 12 + g * 4 + j;
            int bi_ = (i >> (3 - j)) & 1;
            int bk  = (k >> (3 - j)) & 1;
            float wr = ur[u][bi_][bk], wi = ui[u][bi_][bk];
            float nr = pr * wr - pi * wi;
            float ni = pr * wi + pi * wr;
            pr = nr; pi = ni;
        }
        Gr[e] = pr; Gi[e] = pi;
    }
}

// ---------------------------------------------------------------------------
// Kernel 2: full statevector simulation, one workgroup per batch element.
// 256 threads = 8 waves; state (SoA real/imag) lives in LDS (32 KB).
// Fused 16x16 complex gates applied with V_WMMA_F32_16X16X4_F32 chains.
// ---------------------------------------------------------------------------
__global__ __launch_bounds__(256) void qsim(
        const float* __restrict__ x,
        const float* __restrict__ Gr, const float* __restrict__ Gi,
        const float* __restrict__ fcw, const float* __restrict__ fcb,
        float* __restrict__ out) {
    __shared__ float sR[NSTATE];
    __shared__ float sI[NSTATE];
    __shared__ float cs[24];      // cos(x/2)[12], sin(x/2)[12]
    __shared__ float red[8 * 12]; // per-wave partial expvals
    __shared__ float ev[12];

    const int b    = blockIdx.x;
    const int tid  = threadIdx.x;
    const int lane = tid & 31;
    const int wave = tid >> 5;
    const int c    = lane & 15;   // A row / B,C,D column within tile
    const int sel  = lane >> 4;   // lane-half select

    // ---- initial product state: amp[idx] = prod_q (bit ? sin : cos)(x_q/2)
    if (tid < 12) {
        float h = 0.5f * x[b * 12 + tid];
        cs[tid] = cosf(h);
        cs[12 + tid] = sinf(h);
    }
    __syncthreads();
#pragma unroll
    for (int j = 0; j < 16; ++j) {
        int idx = tid + j * 256;
        float p = 1.f;
#pragma unroll
        for (int q = 0; q < 12; ++q)
            p *= cs[q + (((idx >> (11 - q)) & 1) ? 12 : 0)];
        sR[idx] = p;
        sI[idx] = 0.f;
    }
    __syncthreads();

    for (int l = 0; l < N_LAYERS; ++l) {
        // ---- Rot stage: three fused 16x16 complex gates via WMMA ----
#pragma unroll
        for (int g = 0; g < 3; ++g) {
            const float* gr = Gr + (l * 3 + g) * 256;
            const float* gi = Gi + (l * 3 + g) * 256;
            // A tiles (G real/imag), 16x4 per wmma step, K=16 in 4 steps
            v2f ar[4], ai[4];
#pragma unroll
            for (int kk = 0; kk < 4; ++kk) {
                int k0 = kk * 4 + sel * 2;
                v2f t0, t1;
                t0.x = gr[c * 16 + k0]; t0.y = gr[c * 16 + k0 + 1];
                t1.x = gi[c * 16 + k0]; t1.y = gi[c * 16 + k0 + 1];
                ar[kk] = t0; ai[kk] = t1;
            }
            // state element address = row*rs + off  (row = group axis index)
            const int rs = (g == 0) ? 256 : ((g == 1) ? 16 : 1);
#pragma unroll
            for (int tt = 0; tt < 2; ++tt) {
                int tile = wave * 2 + tt;
                int off = (g == 0) ? (tile * 16 + c)
                        : (g == 1) ? (tile * 256 + c)
                                   : (tile * 256 + c * 16);
                // B tiles: 4x16 slices of the 16x16 state block
                v2f br[4], bi_[4];
#pragma unroll
                for (int kk = 0; kk < 4; ++kk) {
                    int k0 = kk * 4 + sel * 2;
                    if (g == 2) {
                        // row-stride 1: (k0, k0+1) are consecutive -> b64 LDS
                        br[kk]  = *(const v2f*)&sR[k0 + off];
                        bi_[kk] = *(const v2f*)&sI[k0 + off];
                    } else {
                        v2f t0, t1;
                        t0.x = sR[k0 * rs + off]; t0.y = sR[(k0 + 1) * rs + off];
                        t1.x = sI[k0 * rs + off]; t1.y = sI[(k0 + 1) * rs + off];
                        br[kk] = t0; bi_[kk] = t1;
                    }
                }
                v8f arr = {}; v8f aii = {}; v8f aim = {};
#pragma unroll
                for (int kk = 0; kk < 4; ++kk)
                    arr = __builtin_amdgcn_wmma_f32_16x16x4_f32(
                        false, ar[kk], false, br[kk], (short)0, arr, false, false);
#pragma unroll
                for (int kk = 0; kk < 4; ++kk)
                    aii = __builtin_amdgcn_wmma_f32_16x16x4_f32(
                        false, ai[kk], false, bi_[kk], (short)0, aii, false, false);
#pragma unroll
                for (int kk = 0; kk < 4; ++kk)
                    aim = __builtin_amdgcn_wmma_f32_16x16x4_f32(
                        false, ar[kk], false, bi_[kk], (short)0, aim, false, false);
#pragma unroll
                for (int kk = 0; kk < 4; ++kk)
                    aim = __builtin_amdgcn_wmma_f32_16x16x4_f32(
                        false, ai[kk], false, br[kk], (short)0, aim, false, false);
                // Re = GrSr - GiSi, Im = GrSi + GiSr
                float dr[8], di[8];
#pragma unroll
                for (int v = 0; v < 8; ++v) { dr[v] = arr[v] - aii[v]; di[v] = aim[v]; }
                // D layout: VGPR v -> row v (lanes 0-15) / v+8 (lanes 16-31)
                if (g == 2) {
                    // row-stride 1: lane writes 8 consecutive floats -> 2x b128
                    int base = off + sel * 8;
                    v4f r0, r1, i0, i1;
                    r0.x=dr[0]; r0.y=dr[1]; r0.z=dr[2]; r0.w=dr[3];
                    r1.x=dr[4]; r1.y=dr[5]; r1.z=dr[6]; r1.w=dr[7];
                    i0.x=di[0]; i0.y=di[1]; i0.z=di[2]; i0.w=di[3];
                    i1.x=di[4]; i1.y=di[5]; i1.z=di[6]; i1.w=di[7];
                    *(v4f*)&sR[base]     = r0;
                    *(v4f*)&sR[base + 4] = r1;
                    *(v4f*)&sI[base]     = i0;
                    *(v4f*)&sI[base + 4] = i1;
                } else {
#pragma unroll
                    for (int v = 0; v < 8; ++v) {
                        int ad = (v + sel * 8) * rs + off;
                        sR[ad] = dr[v];
                        sI[ad] = di[v];
                    }
                }
            }
            __syncthreads();
        }

        // ---- CNOT ring, composed into one permutation gather ----
        const int r = (l % 11) + 1;
        float tr[16], ti[16];
#pragma unroll
        for (int j = 0; j < 16; ++j) {
            int idx = tid + j * 256;
            int src = idx;
#pragma unroll
            for (int q = 11; q >= 0; --q) {            // reverse gate order
                int mc = 1 << (11 - q);
                int mt = 1 << (11 - ((q + r) % 12));
                if (src & mc) src ^= mt;
            }
            tr[j] = sR[src]; ti[j] = sI[src];
        }
        __syncthreads();
#pragma unroll
        for (int j = 0; j < 16; ++j) {
            int idx = tid + j * 256;
            sR[idx] = tr[j]; sI[idx] = ti[j];
        }
        __syncthreads();
    }

    // ---- PauliZ expvals: ev[q] = sum_idx sign(bit_q) * |amp|^2 ----
    float acc[12];
#pragma unroll
    for (int q = 0; q < 12; ++q) acc[q] = 0.f;
#pragma unroll
    for (int j = 0; j < 16; ++j) {
        int idx = tid + j * 256;
        float re = sR[idx], im = sI[idx];
        float p = re * re + im * im;
#pragma unroll
        for (int q = 0; q < 12; ++q)
            acc[q] += ((idx >> (11 - q)) & 1) ? -p : p;
    }
#pragma unroll
    for (int offm = 16; offm > 0; offm >>= 1)
#pragma unroll
        for (int q = 0; q < 12; ++q)
            acc[q] += __shfl_xor(acc[q], offm, 32);
    if (lane == 0)
#pragma unroll
        for (int q = 0; q < 12; ++q) red[wave * 12 + q] = acc[q];
    __syncthreads();
    if (tid < 12) {
        float s = 0.f;
#pragma unroll
        for (int w = 0; w < 8; ++w) s += red[w * 12 + tid];
        ev[tid] = s;
    }
    __syncthreads();

    // ---- linear head ----
    if (tid < N_CLASSES) {
        float s = fcb[tid];
#pragma unroll
        for (int q = 0; q < 12; ++q) s += ev[q] * fcw[tid * 12 + q];
        out[b * N_CLASSES + tid] = s;
    }
}

extern "C" void kernel_launch(void* const* d_in, const int* in_sizes, int n_in,
                              void* d_out, int out_size, void* d_ws, size_t ws_size,
                              hipStream_t stream) {
    const float* x   = (const float*)d_in[0];   // [4096,12]
    const float* qp  = (const float*)d_in[1];   // [6,12,3]
    const float* fcw = (const float*)d_in[2];   // [10,12]
    const float* fcb = (const float*)d_in[3];   // [10]
    float* Gr = (float*)d_ws;                   // 18*256 floats
    float* Gi = Gr + NMAT * 256;                // 18*256 floats (36 KB total)

    prep_gates<<<1, 256, 0, stream>>>(qp, Gr, Gi);
    qsim<<<4096, 256, 0, stream>>>(x, Gr, Gi, fcw, fcb, (float*)d_out);
}